// SparseAttentionPooling_42442866819267
// MI455X (gfx1250) — compile-verified
//
#include <hip/hip_runtime.h>
#include <hip/hip_bf16.h>

// ---------------------------------------------------------------------------
// SparseAttentionPooling for MI455X (gfx1250, wave32)
//
// Sizes (fixed by the reference):
constexpr int Bb = 8;        // batches
constexpr int Nn = 10000;    // nodes
constexpr int Ee = 160000;   // edges
constexpr int Dd = 256;      // feature dim
constexpr int Hh = 128;      // hidden dim
constexpr int Mrows = Bb * Nn;        // 80000 GEMM rows
constexpr float SLOPE = 0.2f;
constexpr int PB = 64;       // partial reduction blocks per batch
constexpr float NEG_INF = -3.402823466e38f;

typedef __attribute__((ext_vector_type(2))) float v2f;
typedef __attribute__((ext_vector_type(8))) float v8f;

// Workspace layout (floats). Total ~21.77M floats (~87.1 MB).
constexpr size_t WS_P      = 0;                          // P [80000,256] (ps|pd)
constexpr size_t WS_SCORES = WS_P + (size_t)Mrows * Dd;  // [B,E]
constexpr size_t WS_PMAX   = WS_SCORES + (size_t)Bb * Ee;  // [B,PB]
constexpr size_t WS_PSUM   = WS_PMAX + Bb * PB;            // [B,PB]
constexpr size_t WS_RMAX   = WS_PSUM + Bb * PB;            // [B]
constexpr size_t WS_RSUM   = WS_RMAX + Bb;                 // [B]
constexpr size_t WS_CNT    = WS_RSUM + Bb;                 // [B]
constexpr size_t WS_ACCUM  = WS_CNT + Bb;                  // [B,2,256]

// ---------------------------------------------------------------------------
// 1) Projection GEMM with f32 WMMA: P[r, 0:128] = X[r,:] @ W1[0:256,:]
//                                   P[r,128:256] = X[r,:] @ W1[256:512,:]
// One wave -> one 16x16 output tile; block = 8 waves covering 128 rows at one
// 16-col tile; the 256x16 weight slab is staged in LDS (16 KB).
__global__ __launch_bounds__(256) void sap_proj_wmma(
    const float* __restrict__ X, const float* __restrict__ W1,
    float* __restrict__ P) {
  __shared__ float sB[256 * 16];
  const int j0 = blockIdx.y * 16;
  const int m0 = blockIdx.x * 128;

  // Stage Wcat[:, j0:j0+16]: col j<128 -> W1[k, j]; col>=128 -> W1[256+k, j-128]
  for (int idx = threadIdx.x; idx < 256 * 16; idx += 256) {
    const int k = idx >> 4, j = idx & 15;
    const int jj = j0 + j;
    sB[idx] = (jj < Hh) ? W1[(size_t)k * Hh + jj]
                        : W1[(size_t)(Dd + k) * Hh + (jj - Hh)];
  }
  __syncthreads();

  const int lane = threadIdx.x & 31;
  const int wave = threadIdx.x >> 5;
  const int mbase = m0 + wave * 16;
  const int mrow  = mbase + (lane & 15);     // A: lanes 0-15 & 16-31 both M=0..15
  const int koff  = (lane >> 4) * 2;         // A: VGPR0=K{0,2}, VGPR1=K{1,3}
  const float* Arow = X + (size_t)mrow * Dd;

  v8f c = {0.f, 0.f, 0.f, 0.f, 0.f, 0.f, 0.f, 0.f};
  #pragma unroll 4
  for (int k0 = 0; k0 < Dd; k0 += 4) {
    const float2 av = *(const float2*)(Arow + k0 + koff);  // global_load_b64
    v2f a; a.x = av.x; a.y = av.y;
    v2f b;
    b.x = sB[(k0 + koff) * 16 + (lane & 15)];
    b.y = sB[(k0 + koff + 1) * 16 + (lane & 15)];
    // (neg_a, A, neg_b, B, c_mod, C, reuse_a, reuse_b)
    c = __builtin_amdgcn_wmma_f32_16x16x4_f32(false, a, false, b, (short)0, c,
                                              false, false);
  }

  // C/D layout: VGPR r -> M = r (+8 for lanes 16-31), N = lane&15
  const int jcol = j0 + (lane & 15);
  const int moff = (lane >> 4) * 8;
  #pragma unroll
  for (int r = 0; r < 8; ++r) {
    P[(size_t)(mbase + r + moff) * Dd + jcol] = c[r];
  }
}

// ---------------------------------------------------------------------------
// 2) Edge scores: one wave per (batch, edge); lane covers 4 of H=128.
__global__ __launch_bounds__(256) void sap_edge_scores(
    const float* __restrict__ P, const int* __restrict__ eidx,
    const float* __restrict__ b1, const float* __restrict__ W2,
    const float* __restrict__ b2, const float* __restrict__ eattr,
    float* __restrict__ scores) {
  const int b = blockIdx.y;
  const int wave = threadIdx.x >> 5, lane = threadIdx.x & 31;
  const int e = blockIdx.x * 8 + wave;           // E = 160000 = 20000*8 exact
  const int s = eidx[e], d = eidx[Ee + e];

  const float4 vs = ((const float4*)(P + (size_t)(b * Nn + s) * Dd))[lane];
  const float4 vd = ((const float4*)(P + (size_t)(b * Nn + d) * Dd + Hh))[lane];
  const float4 vb = ((const float4*)b1)[lane];
  const float4 w2 = ((const float4*)W2)[lane];

  float h0 = vs.x + vd.x + vb.x; h0 = (h0 >= 0.f) ? h0 : SLOPE * h0;
  float h1 = vs.y + vd.y + vb.y; h1 = (h1 >= 0.f) ? h1 : SLOPE * h1;
  float h2 = vs.z + vd.z + vb.z; h2 = (h2 >= 0.f) ? h2 : SLOPE * h2;
  float h3 = vs.w + vd.w + vb.w; h3 = (h3 >= 0.f) ? h3 : SLOPE * h3;
  float acc = h0 * w2.x + h1 * w2.y + h2 * w2.z + h3 * w2.w;

  #pragma unroll
  for (int off = 16; off > 0; off >>= 1) acc += __shfl_xor(acc, off, 32);
  if (lane == 0)
    scores[(size_t)b * Ee + e] = (acc + b2[0]) * eattr[e];
}

// ---------------------------------------------------------------------------
// 3) Softmax: partial max -> reduce -> exp + partial sum -> reduce
__global__ __launch_bounds__(256) void sap_partial_max(
    const float* __restrict__ scores, float* __restrict__ pmax) {
  const int b = blockIdx.y;
  float m = NEG_INF;
  for (int e = blockIdx.x * 256 + threadIdx.x; e < Ee; e += PB * 256)
    m = fmaxf(m, scores[(size_t)b * Ee + e]);
  __shared__ float sm[256];
  sm[threadIdx.x] = m; __syncthreads();
  for (int st = 128; st > 0; st >>= 1) {
    if (threadIdx.x < st) sm[threadIdx.x] = fmaxf(sm[threadIdx.x], sm[threadIdx.x + st]);
    __syncthreads();
  }
  if (threadIdx.x == 0) pmax[b * PB + blockIdx.x] = sm[0];
}

__global__ void sap_reduce64_max(const float* __restrict__ pin, float* __restrict__ pout) {
  __shared__ float s[64];
  s[threadIdx.x] = pin[blockIdx.x * PB + threadIdx.x]; __syncthreads();
  for (int st = 32; st > 0; st >>= 1) {
    if (threadIdx.x < st) s[threadIdx.x] = fmaxf(s[threadIdx.x], s[threadIdx.x + st]);
    __syncthreads();
  }
  if (threadIdx.x == 0) pout[blockIdx.x] = s[0];
}

__global__ __launch_bounds__(256) void sap_exp_psum(
    const float* __restrict__ scores, const float* __restrict__ rowmax,
    float* __restrict__ wout, float* __restrict__ psum) {
  const int b = blockIdx.y;
  const float mx = rowmax[b];
  float acc = 0.f;
  for (int e = blockIdx.x * 256 + threadIdx.x; e < Ee; e += PB * 256) {
    const float v = expf(scores[(size_t)b * Ee + e] - mx);
    wout[(size_t)b * Ee + e] = v;        // unnormalized; normalized in accum pass
    acc += v;
  }
  __shared__ float sm[256];
  sm[threadIdx.x] = acc; __syncthreads();
  for (int st = 128; st > 0; st >>= 1) {
    if (threadIdx.x < st) sm[threadIdx.x] += sm[threadIdx.x + st];
    __syncthreads();
  }
  if (threadIdx.x == 0) psum[b * PB + blockIdx.x] = sm[0];
}

__global__ void sap_reduce64_sum(const float* __restrict__ pin, float* __restrict__ pout) {
  __shared__ float s[64];
  s[threadIdx.x] = pin[blockIdx.x * PB + threadIdx.x]; __syncthreads();
  for (int st = 32; st > 0; st >>= 1) {
    if (threadIdx.x < st) s[threadIdx.x] += s[threadIdx.x + st];
    __syncthreads();
  }
  if (threadIdx.x == 0) pout[blockIdx.x] = s[0];
}

// ---------------------------------------------------------------------------
// 4) mask count per batch
__global__ __launch_bounds__(256) void sap_count_mask(
    const unsigned char* __restrict__ mask, float* __restrict__ cnt) {
  const int b = blockIdx.x;
  float acc = 0.f;
  for (int i = threadIdx.x; i < Nn; i += 256)
    acc += mask[(size_t)b * Nn + i] ? 1.f : 0.f;
  __shared__ float sm[256];
  sm[threadIdx.x] = acc; __syncthreads();
  for (int st = 128; st > 0; st >>= 1) {
    if (threadIdx.x < st) sm[threadIdx.x] += sm[threadIdx.x + st];
    __syncthreads();
  }
  if (threadIdx.x == 0) cnt[b] = sm[0];
}

__global__ void sap_zero_accum(float* __restrict__ accum) {
  accum[blockIdx.x * 256 + threadIdx.x] = 0.f;   // grid = 16 -> 4096 floats
}

// ---------------------------------------------------------------------------
// 5) Weighted edge aggregation (replaces scatter + segment mean):
//    full[b]   += w[b,e] * feats[b, dst[e]]
//    masked[b] += (mask[b,src[e]] ? w[b,e] : 0) * feats[b, dst[e]]
//    Also writes normalized w to d_out.
__global__ __launch_bounds__(256) void sap_edge_accum(
    const float* __restrict__ X, const int* __restrict__ eidx,
    const unsigned char* __restrict__ mask, const float* __restrict__ rowsum,
    float* __restrict__ wout, float* __restrict__ accum) {
  const int b = blockIdx.y;
  const int wave = threadIdx.x >> 5, lane = threadIdx.x & 31;
  const int wi = blockIdx.x * 8 + wave;     // 512 waves per batch
  const float inv = 1.0f / rowsum[b];
  constexpr int STRIDE = PB * 8;            // 512

  float4 f0 = {0,0,0,0}, f1 = {0,0,0,0};    // full-sum, lane covers d=lane*8..+7
  float4 m0 = {0,0,0,0}, m1 = {0,0,0,0};    // masked-sum

  for (int e = wi; e < Ee; e += STRIDE) {
    const size_t widx = (size_t)b * Ee + e;
    const float t = wout[widx] * inv;
    wout[widx] = t;                          // store normalized softmax weight
    const int s = eidx[e], d = eidx[Ee + e];
    const float ms = mask[(size_t)b * Nn + s] ? t : 0.f;

    const float4* F = (const float4*)(X + (size_t)(b * Nn + d) * Dd);
    // prefetch next edge's feature row (CDNA5 global_prefetch_b8)
    if (e + STRIDE < Ee) {
      const int dn = eidx[Ee + e + STRIDE];
      __builtin_prefetch(X + (size_t)(b * Nn + dn) * Dd + lane * 8, 0, 1);
    }
    const float4 a = F[lane * 2];
    const float4 c = F[lane * 2 + 1];
    f0.x += t * a.x; f0.y += t * a.y; f0.z += t * a.z; f0.w += t * a.w;
    f1.x += t * c.x; f1.y += t * c.y; f1.z += t * c.z; f1.w += t * c.w;
    m0.x += ms * a.x; m0.y += ms * a.y; m0.z += ms * a.z; m0.w += ms * a.w;
    m1.x += ms * c.x; m1.y += ms * c.y; m1.z += ms * c.z; m1.w += ms * c.w;
  }

  __shared__ float sF[256], sM[256];
  sF[threadIdx.x] = 0.f; sM[threadIdx.x] = 0.f;
  __syncthreads();
  const int d0 = lane * 8;
  atomicAdd(&sF[d0 + 0], f0.x); atomicAdd(&sF[d0 + 1], f0.y);
  atomicAdd(&sF[d0 + 2], f0.z); atomicAdd(&sF[d0 + 3], f0.w);
  atomicAdd(&sF[d0 + 4], f1.x); atomicAdd(&sF[d0 + 5], f1.y);
  atomicAdd(&sF[d0 + 6], f1.z); atomicAdd(&sF[d0 + 7], f1.w);
  atomicAdd(&sM[d0 + 0], m0.x); atomicAdd(&sM[d0 + 1], m0.y);
  atomicAdd(&sM[d0 + 2], m0.z); atomicAdd(&sM[d0 + 3], m0.w);
  atomicAdd(&sM[d0 + 4], m1.x); atomicAdd(&sM[d0 + 5], m1.y);
  atomicAdd(&sM[d0 + 6], m1.z); atomicAdd(&sM[d0 + 7], m1.w);
  __syncthreads();
  atomicAdd(&accum[(size_t)b * 512 + threadIdx.x], sF[threadIdx.x]);
  atomicAdd(&accum[(size_t)b * 512 + 256 + threadIdx.x], sM[threadIdx.x]);
}

// ---------------------------------------------------------------------------
// 6) graph_feats[b,d] = cnt>0 ? masked/cnt : full/N
__global__ void sap_finalize(const float* __restrict__ accum,
                             const float* __restrict__ cnt,
                             float* __restrict__ out) {
  const int b = blockIdx.x, d = threadIdx.x;
  const float c = cnt[b];
  const float full = accum[(size_t)b * 512 + d];
  const float msk  = accum[(size_t)b * 512 + 256 + d];
  out[(size_t)b * Dd + d] = (c > 0.f) ? (msk / c) : (full / (float)Nn);
}

// ---------------------------------------------------------------------------
extern "C" void kernel_launch(void* const* d_in, const int* in_sizes, int n_in,
                              void* d_out, int out_size, void* d_ws, size_t ws_size,
                              hipStream_t stream) {
  const float* node_feats = (const float*)d_in[0];       // [B,N,D]
  const float* edge_attr  = (const float*)d_in[1];       // [E,1]
  const float* W1         = (const float*)d_in[2];       // [2D,H]
  const float* b1         = (const float*)d_in[3];       // [H]
  const float* W2         = (const float*)d_in[4];       // [H,1]
  const float* b2         = (const float*)d_in[5];       // [1]
  const int*   eidx       = (const int*)d_in[6];         // [2,E] (x64 off -> i32)
  const unsigned char* vm = (const unsigned char*)d_in[7]; // [B,N] bool

  float* ws = (float*)d_ws;
  float* P      = ws + WS_P;
  float* scores = ws + WS_SCORES;
  float* pmax   = ws + WS_PMAX;
  float* psum   = ws + WS_PSUM;
  float* rmax   = ws + WS_RMAX;
  float* rsum   = ws + WS_RSUM;
  float* cnt    = ws + WS_CNT;
  float* accum  = ws + WS_ACCUM;

  float* out_graph = (float*)d_out;            // [B,D]
  float* out_w     = (float*)d_out + Bb * Dd;  // [B,E]

  // 0) zero the atomic accumulators (d_ws is not re-poisoned between replays,
  //    but we must be deterministic every call)
  sap_zero_accum<<<dim3(16), 256, 0, stream>>>(accum);

  // 1) projection GEMM (WMMA f32 16x16x4): 625 row-blocks x 16 col-tiles
  sap_proj_wmma<<<dim3(Mrows / 128, Dd / 16), 256, 0, stream>>>(node_feats, W1, P);

  // 2) edge scores: 8 edges/block, per batch
  sap_edge_scores<<<dim3(Ee / 8, Bb), 256, 0, stream>>>(P, eidx, b1, W2, b2,
                                                        edge_attr, scores);

  // 3) softmax over E per batch
  sap_partial_max<<<dim3(PB, Bb), 256, 0, stream>>>(scores, pmax);
  sap_reduce64_max<<<Bb, 64, 0, stream>>>(pmax, rmax);
  sap_exp_psum<<<dim3(PB, Bb), 256, 0, stream>>>(scores, rmax, out_w, psum);
  sap_reduce64_sum<<<Bb, 64, 0, stream>>>(psum, rsum);

  // 4) mask counts
  sap_count_mask<<<Bb, 256, 0, stream>>>(vm, cnt);

  // 5) normalize w + weighted edge aggregation
  sap_edge_accum<<<dim3(PB, Bb), 256, 0, stream>>>(node_feats, eidx, vm, rsum,
                                                   out_w, accum);

  // 6) finalize graph features
  sap_finalize<<<Bb, Dd, 0, stream>>>(accum, cnt, out_graph);
}